// Conv2dviaLinear_51685636440285
// MI455X (gfx1250) — compile-verified
//
#include <hip/hip_runtime.h>
#include <hip/hip_bf16.h>

// Conv2d-via-giant-sparse-matmul, computed as the underlying dense stride-2
// conv on fp32 WMMA (V_WMMA_F32_16X16X4_F32 keeps exact fp32 semantics).
//
// Dims: x[64,3,64,64], bigW[16,30,30,3,64,64] (structured sparse), bias[14400]
//       out[64,16,30,30]
// bigW[o,0,0,c,r,s] == w0[o,c,r,s], so we read only the dense 16x75 slice.

typedef float v2f __attribute__((ext_vector_type(2)));
typedef float v8f __attribute__((ext_vector_type(8)));

#define B_   64
#define C_   3
#define H_   64
#define W_   64
#define O_   16
#define KW   5
#define STR  2
#define OH_  30
#define OW_  30
#define KRED 75            // C_*KW*KW
#define KPAD 80            // padded to multiple of 4 (K per WMMA)
#define LDSTR 81           // LDS row stride: 81*b mod 64 injective over 16 lanes

__global__ __launch_bounds__(128)
void Conv2dviaLinear_51685636440285_kernel(const float* __restrict__ x,
                                           const float* __restrict__ w,
                                           const float* __restrict__ bias,
                                           float* __restrict__ out) {
    __shared__ float xs[B_ * LDSTR];   // im2col patch: xs[b][k]
    __shared__ float ws[O_ * LDSTR];   // dense kernel: ws[o][k]

    const int ow  = blockIdx.x;
    const int oh  = blockIdx.y;
    const int tid = threadIdx.x;

    // ---- Stage dense 16x75 kernel slice from the structured-sparse big weight.
    // Flat index of big[o,0,0,c,r,s] = o*(OH*OW*C*H*W) + (c*H + r)*W + s
    for (int i = tid; i < O_ * KPAD; i += 128) {
        const int o = i / KPAD, k = i % KPAD;
        float v = 0.0f;
        if (k < KRED) {
            const int c = k / 25, r = (k % 25) / 5, s = k % 5;
            v = w[(size_t)o * ((size_t)OH_ * OW_ * C_ * H_ * W_)
                  + (size_t)(c * H_ + r) * W_ + s];
        }
        ws[o * LDSTR + k] = v;
    }

    // ---- Stage im2col patch for this (oh, ow): xs[b][k] = x[b,c,2oh+r,2ow+s]
    const int ih0 = oh * STR;
    const int iw0 = ow * STR;
    for (int i = tid; i < B_ * KPAD; i += 128) {
        const int b = i / KPAD, k = i % KPAD;
        float v = 0.0f;
        if (k < KRED) {
            const int c = k / 25, r = (k % 25) / 5, s = k % 5;
            v = x[((b * C_ + c) * H_ + (ih0 + r)) * W_ + (iw0 + s)];
        }
        xs[b * LDSTR + k] = v;
    }

    __syncthreads();

    // ---- One wave per 16-row b-tile: [16 x 16 x 80] fp32 GEMM on WMMA.
    const int wid   = tid >> 5;          // 0..3 : b-tile
    const int lane  = tid & 31;
    const int m     = lane & 15;
    const int khalf = (lane < 16) ? 0 : 2;   // A/B VGPR0 holds K=0(lo)/K=2(hi)

    const float* xrow = &xs[(wid * 16 + m) * LDSTR];  // A: row = b, striped by K
    const float* wrow = &ws[m * LDSTR];               // B: col = o (lane), striped by K

    v8f acc = {0.f, 0.f, 0.f, 0.f, 0.f, 0.f, 0.f, 0.f};
#pragma unroll
    for (int step = 0; step < KPAD / 4; ++step) {
        const int k0 = step * 4 + khalf;
        v2f a, bm;
        a.x  = xrow[k0];      // VGPR0: K = k0   (lanes<16: +0, lanes>=16: +2)
        a.y  = xrow[k0 + 1];  // VGPR1: K = k0+1 (lanes<16: +1, lanes>=16: +3)
        bm.x = wrow[k0];
        bm.y = wrow[k0 + 1];
        // 8 args: (neg_a, A, neg_b, B, c_mod, C, reuse_a, reuse_b)
        acc = __builtin_amdgcn_wmma_f32_16x16x4_f32(
                  false, a, false, bm, (short)0, acc, false, false);
    }

    // ---- Epilogue: D layout -> VGPR v: M = v + (lane>=16 ? 8 : 0), N = lane%16
    const int o     = m;
    const int pos   = oh * OW_ + ow;
    const float bb  = bias[o * (OH_ * OW_) + pos];
    const int mbase = wid * 16 + ((lane < 16) ? 0 : 8);
#pragma unroll
    for (int v = 0; v < 8; ++v) {
        const int b = mbase + v;
        out[(size_t)b * (O_ * OH_ * OW_) + (size_t)o * (OH_ * OW_) + pos] =
            acc[v] + bb;
    }
}

extern "C" void kernel_launch(void* const* d_in, const int* in_sizes, int n_in,
                              void* d_out, int out_size, void* d_ws, size_t ws_size,
                              hipStream_t stream) {
    (void)in_sizes; (void)n_in; (void)out_size; (void)d_ws; (void)ws_size;
    const float* x    = (const float*)d_in[0];
    const float* w    = (const float*)d_in[1];
    const float* bias = (const float*)d_in[2];
    float*       out  = (float*)d_out;

    dim3 grid(OW_, OH_);   // one block per output spatial position
    Conv2dviaLinear_51685636440285_kernel<<<grid, dim3(128), 0, stream>>>(x, w, bias, out);
}